// TransducerJoint_74045236183253
// MI455X (gfx1250) — compile-verified
//
#include <hip/hip_runtime.h>
#include <hip/hip_bf16.h>
#include <math.h>

typedef __attribute__((ext_vector_type(2))) float v2f;
typedef __attribute__((ext_vector_type(8))) float v8f;

#define BT  4
#define TT  200
#define UU  100
#define EDIM 512
#define JDIM 640
#define VOCAB 1024
#define TU   (TT * UU)            // 20000
#define MROWS (BT * TT * UU)      // 80000 rows of final GEMM

static __device__ __forceinline__ v8f wmma_f32(v2f a, v2f b, v8f c) {
    return __builtin_amdgcn_wmma_f32_16x16x4_f32(false, a, false, b, (short)0, c,
                                                 false, false);
}

// ---------------------------------------------------------------------------
// Kernel 2 (emitted FIRST so its disasm/metadata shows in the snippet):
// fused tanh-join + vocab GEMM.
// Workgroup: 512 threads (16 waves), 64 output rows x 1024 cols.
// Phase 1: LDS aTile[64][640] = tanh(encP[b,t,:] + predP[b,u,:])
//          row stride padded to 644 floats -> conflict-free ds_load_b64.
// Phase 2: wave w computes C[64 x 64] at cols [w*64, w*64+64):
//          per K-step of 4: 4 LDS A b64 loads, 8 global b32 B loads (L2-hot),
//          16 x v_wmma_f32_16x16x4_f32 (B reused across 4 M sub-tiles).
// Output (328 MB write-once > 192 MB L2) stored non-temporal.
// ---------------------------------------------------------------------------
#define LSTR  644
#define MTILE 64

__global__ __launch_bounds__(512) void joint_kernel(
    const float* __restrict__ encP, const float* __restrict__ predP,
    const float* __restrict__ W_out, const float* __restrict__ b_out,
    float* __restrict__ out)
{
    __shared__ float aTile[MTILE * LSTR];        // 164,864 bytes

    const int tid  = threadIdx.x;
    const int row0 = blockIdx.x * MTILE;         // 1250 blocks cover 80000 rows

    // -------- Phase 1: build 64 joint rows in LDS --------
    for (int i = tid; i < MTILE * JDIM; i += 512) {
        const int r = i / JDIM;
        const int j = i - r * JDIM;
        const int m = row0 + r;
        const int b   = m / TU;
        const int rem = m - b * TU;
        const int t   = rem / UU;
        const int u   = rem - t * UU;
        const float e = encP[((size_t)b * TT + t) * JDIM + j];
        const float p = predP[((size_t)b * UU + u) * JDIM + j];
        aTile[r * LSTR + j] = tanhf(e + p);
    }
    __syncthreads();

    // -------- Phase 2: WMMA GEMM against W_out --------
    const int wave  = tid >> 5;
    const int lane  = tid & 31;
    const int n16   = lane & 15;
    const int khalf = lane >> 4;
    const int col0  = wave * 64;

    const float* aP = &aTile[n16 * LSTR + 2 * khalf];
    const float* bP = W_out + (size_t)(2 * khalf) * VOCAB + col0 + n16;

    v8f c[4][4];
#pragma unroll
    for (int rg = 0; rg < 4; ++rg)
#pragma unroll
        for (int cg = 0; cg < 4; ++cg)
            c[rg][cg] = (v8f){};

    for (int k = 0; k < JDIM; k += 4) {
        v2f a[4];
#pragma unroll
        for (int rg = 0; rg < 4; ++rg)                       // rows rg*16..rg*16+15
            a[rg] = *(const v2f*)(aP + rg * 16 * LSTR + k);
        const float* bk0 = bP + (size_t)k * VOCAB;           // row k + 2*khalf
        const float* bk1 = bk0 + VOCAB;                      // row k + 2*khalf + 1
        v2f b[4];
#pragma unroll
        for (int cg = 0; cg < 4; ++cg)
            b[cg] = (v2f){ bk0[16 * cg], bk1[16 * cg] };
#pragma unroll
        for (int rg = 0; rg < 4; ++rg)
#pragma unroll
            for (int cg = 0; cg < 4; ++cg)
                c[rg][cg] = wmma_f32(a[rg], b[cg], c[rg][cg]);
    }

    float bias[4];
#pragma unroll
    for (int cg = 0; cg < 4; ++cg)
        bias[cg] = b_out[col0 + 16 * cg + n16];

#pragma unroll
    for (int rg = 0; rg < 4; ++rg) {
#pragma unroll
        for (int r = 0; r < 8; ++r) {
            const int m = rg * 16 + r + 8 * khalf;           // C/D layout
            float* o = out + (size_t)(row0 + m) * VOCAB + col0 + n16;
#pragma unroll
            for (int cg = 0; cg < 4; ++cg)
                __builtin_nontemporal_store(c[rg][cg][r] + bias[cg], o + 16 * cg);
        }
    }
}

// ---------------------------------------------------------------------------
// Kernel 1: fused projection GEMM.
//   rows 0..799   : encP = enc_out[800,512] @ W_enc[512,640] + b_enc
//   rows 800..1199: predP = pred_out[400,512] @ W_pred[512,640]
// One 16x16 tile per wave, K=512 in steps of 4 via v_wmma_f32_16x16x4_f32.
// Grid: 75 M-tiles * 40 N-tiles = 3000 wave-tiles = 375 blocks of 8 waves.
// ---------------------------------------------------------------------------
__global__ __launch_bounds__(256) void proj_kernel(
    const float* __restrict__ enc_out, const float* __restrict__ pred_out,
    const float* __restrict__ W_enc, const float* __restrict__ b_enc,
    const float* __restrict__ W_pred,
    float* __restrict__ encP, float* __restrict__ predP)
{
    const int wave  = threadIdx.x >> 5;
    const int lane  = threadIdx.x & 31;
    const int tile  = blockIdx.x * 8 + wave;     // 0..2999
    const int mt    = tile / 40;                 // 0..74
    const int nt    = tile - mt * 40;            // 0..39
    const int row0  = mt * 16;
    const int col0  = nt * 16;
    const int n16   = lane & 15;
    const int khalf = lane >> 4;                 // 0 or 1

    const bool isEnc = (row0 < BT * TT);
    const float* __restrict__ A =
        isEnc ? (enc_out + (size_t)row0 * EDIM)
              : (pred_out + (size_t)(row0 - BT * TT) * EDIM);
    const float* __restrict__ W = isEnc ? W_enc : W_pred;
    float* __restrict__ Dst =
        isEnc ? (encP + (size_t)row0 * JDIM)
              : (predP + (size_t)(row0 - BT * TT) * JDIM);

    // A fragment: lane holds A[m = lane%16][k + 2*khalf + {0,1}] (float2)
    const float* aP = A + (size_t)n16 * EDIM + 2 * khalf;
    // B fragment: lane holds W[k + 2*khalf + {0,1}][col0 + n16]
    const float* bP = W + (size_t)(2 * khalf) * JDIM + col0 + n16;

    v8f acc = {};
    for (int k = 0; k < EDIM; k += 4) {
        v2f a = *(const v2f*)(aP + k);
        v2f b;
        b.x = bP[(size_t)k * JDIM];
        b.y = bP[(size_t)(k + 1) * JDIM];
        acc = wmma_f32(a, b, acc);
    }

    const float bias = isEnc ? b_enc[col0 + n16] : 0.0f;
#pragma unroll
    for (int r = 0; r < 8; ++r) {
        const int m = r + 8 * khalf;             // C/D layout: VGPR r -> M = r (+8)
        Dst[(size_t)m * JDIM + col0 + n16] = acc[r] + bias;
    }
}

// ---------------------------------------------------------------------------
extern "C" void kernel_launch(void* const* d_in, const int* in_sizes, int n_in,
                              void* d_out, int out_size, void* d_ws, size_t ws_size,
                              hipStream_t stream) {
    const float* enc_out  = (const float*)d_in[0];   // [4,200,512]
    const float* pred_out = (const float*)d_in[1];   // [4,100,512]
    const float* W_enc    = (const float*)d_in[2];   // [512,640]
    const float* b_enc    = (const float*)d_in[3];   // [640]
    const float* W_pred   = (const float*)d_in[4];   // [512,640]
    const float* W_out    = (const float*)d_in[5];   // [640,1024]
    const float* b_out    = (const float*)d_in[6];   // [1024]
    float* out = (float*)d_out;                      // [4,200,100,1024]

    float* encP  = (float*)d_ws;                     // [800,640]
    float* predP = encP + (size_t)(BT * TT) * JDIM;  // [400,640]

    // 1200x512x640 projection GEMM: 3000 wave-tiles / 8 waves per block
    hipLaunchKernelGGL(proj_kernel, dim3(375), dim3(256), 0, stream,
                       enc_out, pred_out, W_enc, b_enc, W_pred, encP, predP);

    // 80000x640x1024 fused tanh-join GEMM: 64 rows per workgroup
    hipLaunchKernelGGL(joint_kernel, dim3(MROWS / MTILE), dim3(512), 0, stream,
                       encP, predP, W_out, b_out, out);
}